// MultiViewNetwork_76295799046548
// MI455X (gfx1250) — compile-verified
//
#include <hip/hip_runtime.h>
#include <math.h>

namespace {
constexpr int B = 8, C = 32;
constexpr int Hb = 256, Wb = 256;
constexpr int Hr = 32, Wr = 1024;
constexpr int HWb = Hb * Wb;      // 65536
constexpr int HWr = Hr * Wr;      // 32768
constexpr int OPC = HWr + HWb;    // 98304 elements per (b, c) in the flat output
constexpr float NEGF = -1.0e9f;

// Mirror Python: math.radians(x) = x * (pi/180) in double, then cast to f32 at use.
constexpr double DEG_D     = M_PI / 180.0;
constexpr double PHI_MIN_D = -M_PI;
constexpr double PHI_MAX_D =  M_PI;
constexpr double TH_MIN_D  = -25.0 * DEG_D;
constexpr double TH_MAX_D  =   3.0 * DEG_D;
constexpr float  PHI_MINF  = (float)PHI_MIN_D;
constexpr float  PHI_DENF  = (float)(PHI_MAX_D - PHI_MIN_D);
constexpr float  TH_MAXF   = (float)TH_MAX_D;
constexpr float  TH_DENF   = (float)(TH_MAX_D - TH_MIN_D);
} // namespace

__device__ __forceinline__ void atomic_max_f32(float* addr, float val) {
    // Confirmed (round 1): lowers to single no-return global_atomic_max_num_f32.
    __hip_atomic_fetch_max(addr, val, __ATOMIC_RELAXED, __HIP_MEMORY_SCOPE_AGENT);
}

// ---------------------------------------------------------------------------
// Initialize the whole output to -1e9 with 128-bit stores. Runs first on the
// stream every call (harness poisons d_out). Output (96MB) stays resident in
// the 192MB L2, so the subsequent atomic streams are L2-local.
// ---------------------------------------------------------------------------
__global__ __launch_bounds__(256) void k_init(float4* __restrict__ out, int n4) {
    int i = blockIdx.x * 256 + threadIdx.x;
    if (i < n4) out[i] = make_float4(NEGF, NEGF, NEGF, NEGF);
}

// ---------------------------------------------------------------------------
// BEV (des) -> spherical: one thread per (b, bev_pixel).
// idx computed once; all 32 channel values staged into registers (32 loads in
// flight, LOADcnt<=32<63), then 32 hardware fmax atomics fired back-to-back.
// Grid = B*HWb/256 exactly (no tail).
// ---------------------------------------------------------------------------
__global__ __launch_bounds__(256) void k_des2sph(const float* __restrict__ bev_feat,
                                                 const float* __restrict__ z_map,
                                                 float* __restrict__ out) {
#pragma clang fp contract(off)
    const int t = blockIdx.x * 256 + threadIdx.x;   // [0, B*HWb)
    const int b = t >> 16;                          // / 65536
    const int p = t & (HWb - 1);
    const int v = p >> 8;                           // row
    const int u = p & (Wb - 1);                     // col

    const float* __restrict__ src = bev_feat + (size_t)b * C * HWb + p;
    __builtin_prefetch(src, 0, 3);                  // global_prefetch_b8: warm L2

    const float x = -50.0f + ((float)u + 0.5f) / 256.0f * 100.0f;
    const float y =  50.0f - ((float)v + 0.5f) / 256.0f * 100.0f;
    const float z = z_map[t];
    const float r = sqrtf(x * x + y * y + z * z) + 1e-6f;
    const float phi   = atan2f(y, x);
    const float theta = asinf(z / r);

    const int j = (int)((phi - PHI_MINF) / PHI_DENF * (float)(Wr - 1));
    const int i = (int)((TH_MAXF - theta) / TH_DENF * (float)(Hr - 1));
    int idx = i * Wr + j;
    idx = idx < 0 ? 0 : (idx > HWr - 1 ? HWr - 1 : idx);

    float* __restrict__ dst = out + (size_t)b * C * OPC + idx;

    float vals[C];
#pragma unroll
    for (int c = 0; c < C; ++c)
        vals[c] = src[(size_t)c * HWb];             // coalesced; all in flight
#pragma unroll
    for (int c = 0; c < C; ++c)
        atomic_max_f32(dst + (size_t)c * OPC, vals[c]);  // global_atomic_max_num_f32
}

// ---------------------------------------------------------------------------
// Spherical -> BEV (des): one thread per (b, sph_pixel). Grid = B*HWr/256.
// ---------------------------------------------------------------------------
__global__ __launch_bounds__(256) void k_sph2des(const float* __restrict__ rv_feat,
                                                 const float* __restrict__ r_map,
                                                 float* __restrict__ out) {
#pragma clang fp contract(off)
    const int t = blockIdx.x * 256 + threadIdx.x;   // [0, B*HWr)
    const int b = t >> 15;                          // / 32768
    const int p = t & (HWr - 1);
    const int i = p >> 10;                          // theta row
    const int j = p & (Wr - 1);                     // phi col

    const float* __restrict__ src = rv_feat + (size_t)b * C * HWr + p;
    __builtin_prefetch(src, 0, 3);                  // global_prefetch_b8: warm L2

    const float phi   = PHI_MINF + (float)j / (float)(Wr - 1) * PHI_DENF;
    const float theta = TH_MAXF - (float)i / (float)(Hr - 1) * TH_DENF;
    const float r  = r_map[t];
    const float ct = cosf(theta);
    const float x  = r * ct * cosf(phi);
    const float y  = r * ct * sinf(phi);

    const int u = (int)((x + 50.0f) / 100.0f * (float)(Wb - 1));
    const int v = (int)((50.0f - y) / 100.0f * (float)(Hb - 1));
    int idx = v * Wb + u;
    idx = idx < 0 ? 0 : (idx > HWb - 1 ? HWb - 1 : idx);

    float* __restrict__ dst = out + (size_t)b * C * OPC + HWr + idx;

    float vals[C];
#pragma unroll
    for (int c = 0; c < C; ++c)
        vals[c] = src[(size_t)c * HWr];             // coalesced; all in flight
#pragma unroll
    for (int c = 0; c < C; ++c)
        atomic_max_f32(dst + (size_t)c * OPC, vals[c]);  // global_atomic_max_num_f32
}

extern "C" void kernel_launch(void* const* d_in, const int* in_sizes, int n_in,
                              void* d_out, int out_size, void* d_ws, size_t ws_size,
                              hipStream_t stream) {
    const float* bev_feat = (const float*)d_in[0];  // (8,32,256,256)
    const float* z_map    = (const float*)d_in[1];  // (8,1,256,256)
    const float* rv_feat  = (const float*)d_in[2];  // (8,32,32,1024)
    const float* r_map    = (const float*)d_in[3];  // (8,1,32,1024)
    float* out = (float*)d_out;                     // (8,32,98304) flat

    const int n4 = out_size / 4;                    // out_size divisible by 4
    k_init<<<(n4 + 255) / 256, 256, 0, stream>>>((float4*)out, n4);
    k_des2sph<<<(B * HWb) / 256, 256, 0, stream>>>(bev_feat, z_map, out);
    k_sph2des<<<(B * HWr) / 256, 256, 0, stream>>>(rv_feat, r_map, out);
}